// Attention_26774826124067
// MI455X (gfx1250) — compile-verified
//
#include <hip/hip_runtime.h>

// CDNA5 / gfx1250, wave32. Exploits the reference's algebraic identity:
// softmax rows sum to 1  =>  z == V  =>  out = x @ (WVf @ WOf) + (bVf @ WOf + bO).
// Two fp32 WMMA GEMMs (V_WMMA_F32_16X16X4_F32) staged through LDS.
//   - block tile 256x128, 8 waves as 4x2 grid of 64x64 wave tiles
//   - both LDS tiles stored [outer][K+pad] so every fragment is one ds_load_b64
//   - register-prefetch double buffering: next stage's global loads overlap WMMA
// Workspace: W_c (2048*2048 f32 = 16 MB) + combined bias (2048 f32).

typedef __attribute__((ext_vector_type(2))) float v2f;
typedef __attribute__((ext_vector_type(8))) float v8f;

#define TILE_M 256
#define TILE_N 128
#define TILE_K 16
#define LSTRIDE 20   // 16 + 4 pad: 20*L mod 64 -> distinct multiples of 4 => conflict-free b64 reads

// Generic fp32 GEMM: C[M,N] = A[M,K] * B[K,N] (+ bias[N] if bias != nullptr)
// A element (row,k) is at A[(k>>7)*aBlockStride + row*aRowStride + (k&127)]:
//   - plain row-major:  aRowStride = K, aBlockStride = 128
//   - W_V [H,D,DH] viewed as [D, H*DH]: aRowStride = DH(=128), aBlockStride = D*DH
__global__ __launch_bounds__(256)
void gemm_f32_wmma(const float* __restrict__ A,
                   const float* __restrict__ B,
                   float* __restrict__ C,
                   const float* __restrict__ bias,
                   int N, int K,
                   long aRowStride, long aBlockStride)
{
    __shared__ float As[TILE_M * LSTRIDE];  // [m][k] rows of 16 + pad
    __shared__ float Bt[TILE_N * LSTRIDE];  // [n][k] transposed tile, rows of 16 + pad

    const int t     = threadIdx.x;
    const int lane  = t & 31;
    const int wave  = t >> 5;       // 0..7
    const int waveM = wave >> 1;    // 0..3 -> 64-row strip
    const int waveN = wave & 1;     // 0..1 -> 64-col strip
    const int lh    = lane >> 4;    // lane half: f32 frag K split {0,1} vs {2,3}
    const int ll    = lane & 15;

    const int tileN0 = blockIdx.x * TILE_N;
    const int tileM0 = blockIdx.y * TILE_M;

    // B staging: one column per thread (coalesced: 128 consecutive threads = one row)
    const int bcol = t & 127;
    const int brh  = (t >> 7) * 8;  // row-half of the 16-deep K slab

    const v8f vzero = {0.f, 0.f, 0.f, 0.f, 0.f, 0.f, 0.f, 0.f};
    v8f acc[4][4];
#pragma unroll
    for (int i = 0; i < 4; ++i)
#pragma unroll
        for (int j = 0; j < 4; ++j)
            acc[i][j] = vzero;

    float4 pa[4];
    float  pb[8];

    auto load_stage = [&](int kb) {
#pragma unroll
        for (int r = 0; r < 4; ++r) {            // A tile: 256x16 = 1024 float4
            const int idx  = t + r * 256;
            const int arow = idx >> 2;            // 0..255
            const int ac4  = (idx & 3) << 2;      // 0,4,8,12
            const int gk   = kb + ac4;
            const long aoff = (long)(gk >> 7) * aBlockStride
                            + (long)(tileM0 + arow) * aRowStride
                            + (gk & 127);
            pa[r] = *(const float4*)(A + aoff);
        }
#pragma unroll
        for (int q = 0; q < 8; ++q)              // B tile: 16x128, column strip per thread
            pb[q] = B[(long)(kb + brh + q) * N + tileN0 + bcol];
    };

    load_stage(0);

    for (int kb = 0; kb < K; kb += TILE_K) {
        __syncthreads();   // previous stage's LDS readers are done
#pragma unroll
        for (int r = 0; r < 4; ++r) {
            const int idx = t + r * 256;
            *(float4*)(&As[(idx >> 2) * LSTRIDE + ((idx & 3) << 2)]) = pa[r];
        }
        *(float4*)(&Bt[bcol * LSTRIDE + brh])     = make_float4(pb[0], pb[1], pb[2], pb[3]);
        *(float4*)(&Bt[bcol * LSTRIDE + brh + 4]) = make_float4(pb[4], pb[5], pb[6], pb[7]);
        __syncthreads();

        if (kb + TILE_K < K) load_stage(kb + TILE_K);   // overlap with WMMA below

#pragma unroll
        for (int kk = 0; kk < TILE_K; kk += 4) {
            v2f a[4], b[4];
#pragma unroll
            for (int i = 0; i < 4; ++i)   // A frag (16x4 f32): lanes 0-15 K={kk,kk+1}, 16-31 K={kk+2,kk+3}
                a[i] = *(const v2f*)(&As[(waveM * 64 + i * 16 + ll) * LSTRIDE + kk + 2 * lh]);
#pragma unroll
            for (int j = 0; j < 4; ++j)   // B frag: same pairing, read from transposed tile
                b[j] = *(const v2f*)(&Bt[(waveN * 64 + j * 16 + ll) * LSTRIDE + kk + 2 * lh]);
#pragma unroll
            for (int i = 0; i < 4; ++i)
#pragma unroll
                for (int j = 0; j < 4; ++j)
                    acc[i][j] = __builtin_amdgcn_wmma_f32_16x16x4_f32(
                        false, a[i], false, b[j], (short)0, acc[i][j], false, false);
        }
    }

    // Epilogue: C/D layout = VGPR r -> rows r (lanes 0-15) / r+8 (lanes 16-31)
#pragma unroll
    for (int j = 0; j < 4; ++j) {
        const int col = tileN0 + waveN * 64 + j * 16 + ll;
        const float bv = bias ? bias[col] : 0.f;
#pragma unroll
        for (int i = 0; i < 4; ++i) {
            const int rowBase = tileM0 + waveM * 64 + i * 16 + lh * 8;
#pragma unroll
            for (int r = 0; r < 8; ++r)
                C[(long)(rowBase + r) * N + col] = acc[i][j][r] + bv;
        }
    }
}

// bcomb[d] = b_O[d] + sum_f bV_flat[f] * W_O_flat[f, d]
__global__ __launch_bounds__(256)
void bias_combine(const float* __restrict__ bV, const float* __restrict__ WO,
                  const float* __restrict__ bO, float* __restrict__ bcomb,
                  int F, int Dm)
{
    const int d = blockIdx.x * blockDim.x + threadIdx.x;
    if (d >= Dm) return;
    float s = bO[d];
    for (int f = 0; f < F; ++f)
        s = fmaf(bV[f], WO[(long)f * Dm + d], s);
    bcomb[d] = s;
}

extern "C" void kernel_launch(void* const* d_in, const int* in_sizes, int n_in,
                              void* d_out, int out_size, void* d_ws, size_t ws_size,
                              hipStream_t stream) {
    (void)in_sizes; (void)n_in; (void)out_size; (void)ws_size;

    const float* x  = (const float*)d_in[0];   // [B,S,D] = [8192, 2048]
    const float* WV = (const float*)d_in[5];   // [H,D,DH]
    const float* bV = (const float*)d_in[6];   // [H,DH] -> flat [2048]
    const float* WO = (const float*)d_in[7];   // [H,DH,D] == row-major [2048, 2048]
    const float* bO = (const float*)d_in[8];   // [D]
    float* out = (float*)d_out;                // [8192, 2048]

    const int D = 2048, DH = 128, H = 16, BS = 8192;

    float* Wc    = (float*)d_ws;                       // D*D floats (16 MB)
    float* bcomb = (float*)d_ws + (size_t)D * D;       // D floats

    // Fold biases: bcomb = bV_flat @ WO_flat + bO
    bias_combine<<<(D + 255) / 256, 256, 0, stream>>>(bV, WO, bO, bcomb, H * DH, D);

    // GEMM1: Wc[d',d] = sum_f WV_flat[d',f] * WO_flat[f,d]   (2048^3)
    {
        dim3 grid(D / TILE_N, D / TILE_M);
        gemm_f32_wmma<<<grid, 256, 0, stream>>>(WV, WO, Wc, nullptr,
                                                D, H * DH,
                                                /*aRowStride=*/(long)DH,
                                                /*aBlockStride=*/(long)D * DH);
    }

    // GEMM2: out = x @ Wc + bcomb   (8192 x 2048 x 2048)
    {
        dim3 grid(D / TILE_N, BS / TILE_M);
        gemm_f32_wmma<<<grid, 256, 0, stream>>>(x, Wc, out, bcomb,
                                                D, D,
                                                /*aRowStride=*/(long)D,
                                                /*aBlockStride=*/128);
    }
}